// GraphAttentionLayer2_91036126806379
// MI455X (gfx1250) — compile-verified
//
#include <hip/hip_runtime.h>
#include <math.h>

#define B_    4
#define N_    4096
#define E_    65536
#define FIN_  128
#define FOUT_ 64

typedef __attribute__((ext_vector_type(2))) float v2f;
typedef __attribute__((ext_vector_type(8))) float v8f;

// ---------------------------------------------------------------------------
// K1: Wh[B*N, 64] = h[B*N, 128] @ W[128, 64] using V_WMMA_F32_16X16X4_F32.
// One wave computes one 16x16 output tile; block = 4 waves = the 4 N-tiles.
// A (16x4 f32): lanes 0-15 hold M=lane, {K, K+1}; lanes 16-31 hold {K+2, K+3}.
// B (4x16 f32): lanes 0-15 hold N=lane, rows {K, K+1}; lanes 16-31 rows {K+2,K+3}.
// C/D (16x16 f32): VGPR r -> M = r + (lane<16 ? 0 : 8), N = lane%16.
// ---------------------------------------------------------------------------
__global__ void gat_wh_gemm(const float* __restrict__ h,
                            const float* __restrict__ W,
                            float* __restrict__ Wh) {
  const int lane  = threadIdx.x & 31;
  const int ntile = threadIdx.x >> 5;          // 0..3
  const int m0    = blockIdx.x << 4;
  const int n0    = ntile << 4;
  const int row   = m0 + (lane & 15);
  const int col   = n0 + (lane & 15);
  const int khalf = (lane >> 4) << 1;          // 0 or 2

  v8f c = {};
  #pragma unroll
  for (int k = 0; k < FIN_; k += 4) {
    v2f a, b;
    const float* ap = h + (size_t)row * FIN_ + k + khalf;
    a.x = ap[0];
    a.y = ap[1];
    const float* bp = W + (size_t)(k + khalf) * FOUT_ + col;
    b.x = bp[0];
    b.y = bp[FOUT_];
    c = __builtin_amdgcn_wmma_f32_16x16x4_f32(false, a, false, b,
                                              (short)0, c, false, false);
  }

  const int mbase = (lane >> 4) << 3;          // 0 or 8
  float* out = Wh + (size_t)(m0 + mbase) * FOUT_ + col;
  #pragma unroll
  for (int r = 0; r < 8; ++r) out[(size_t)r * FOUT_] = c[r];
}

// ---------------------------------------------------------------------------
// K2: per-node scalars from batch-0 rows of Wh (faithful flat-gather quirk):
//   s1[n] = Wh0[n] . a[0:64],  s2[n] = Wh0[n] . a[64:128]
// ---------------------------------------------------------------------------
__global__ void gat_node_scores(const float* __restrict__ Wh,
                                const float* __restrict__ a,
                                float* __restrict__ s1,
                                float* __restrict__ s2) {
  int n = blockIdx.x * blockDim.x + threadIdx.x;
  if (n >= N_) return;
  const float* row = Wh + (size_t)n * FOUT_;
  float x1 = 0.f, x2 = 0.f;
  #pragma unroll
  for (int f = 0; f < FOUT_; ++f) {
    x1 += row[f] * a[f];
    x2 += row[f] * a[FOUT_ + f];
  }
  s1[n] = x1;
  s2[n] = x2;
}

// ---------------------------------------------------------------------------
// Pass A: dedup edges (bitmap, first-setter wins == last-write-wins since the
// scattered value depends only on (src,tgt)), append distinct positive edges
// to a compact list, and track per-column max via bit-pattern atomicMax
// (valid: all kept e > 0, colMax initialized to 0x00000000).
// ---------------------------------------------------------------------------
__global__ void gat_edge_pass_a(const int* __restrict__ edge,
                                const float* __restrict__ s1,
                                const float* __restrict__ s2,
                                unsigned* __restrict__ bitmap,
                                float* __restrict__ colMax,
                                unsigned* __restrict__ cnt,
                                unsigned long long* __restrict__ list) {
  int gid = blockIdx.x * blockDim.x + threadIdx.x;
  if (gid >= B_ * E_) return;
  int b = gid / E_, k = gid % E_;
  int src = edge[(size_t)b * 2 * E_ + k];
  int tgt = edge[(size_t)b * 2 * E_ + E_ + k];
  float e = s1[src] + s2[tgt];
  if (e > 0.f) {
    unsigned long long bitidx =
        ((unsigned long long)b * N_ + (unsigned)src) * N_ + (unsigned)tgt;
    unsigned mask = 1u << (unsigned)(bitidx & 31);
    unsigned old  = atomicOr(&bitmap[bitidx >> 5], mask);
    if (!(old & mask)) {
      unsigned pos = atomicAdd(cnt, 1u);
      unsigned key = ((unsigned)b << 24) | ((unsigned)src << 12) | (unsigned)tgt;
      list[pos] = ((unsigned long long)__float_as_uint(e) << 32) | key;
      atomicMax((unsigned*)&colMax[b * N_ + tgt], __float_as_uint(e));
    }
  }
}

// Pass B: column softmax denominators over distinct positive entries.
// (NEG entries contribute exp(-9e15 - max) == 0.0f exactly in f32.)
__global__ void gat_edge_pass_b(const unsigned long long* __restrict__ list,
                                const unsigned* __restrict__ cnt,
                                const float* __restrict__ colMax,
                                float* __restrict__ colSum) {
  unsigned gid = blockIdx.x * blockDim.x + threadIdx.x;
  if (gid >= *cnt) return;
  unsigned long long v = list[gid];
  unsigned key = (unsigned)v;
  float e = __uint_as_float((unsigned)(v >> 32));
  int b = key >> 24, tgt = key & 0xFFF;
  int cidx = b * N_ + tgt;
  atomicAdd(&colSum[cidx], expf(e - colMax[cidx]));
}

// Pass C: sparse part of attention @ Wh: accum[b,src,:] += w * Wh[b,tgt,:]
__global__ void gat_edge_pass_c(const unsigned long long* __restrict__ list,
                                const unsigned* __restrict__ cnt,
                                const float* __restrict__ colMax,
                                const float* __restrict__ colSum,
                                const float* __restrict__ Wh,
                                float* __restrict__ accum) {
  unsigned gid   = blockIdx.x * blockDim.x + threadIdx.x;
  unsigned entry = gid >> 6;
  int f = gid & 63;
  if (entry >= *cnt) return;
  unsigned long long v = list[entry];
  unsigned key = (unsigned)v;
  float e = __uint_as_float((unsigned)(v >> 32));
  int b = key >> 24, src = (key >> 12) & 0xFFF, tgt = key & 0xFFF;
  int cidx = b * N_ + tgt;
  float w = expf(e - colMax[cidx]) / colSum[cidx];
  atomicAdd(&accum[((size_t)b * N_ + src) * FOUT_ + f],
            w * Wh[((size_t)b * N_ + tgt) * FOUT_ + f]);
}

// Pass D: rank-1 uniform term from columns with no positive entry:
//   u[b,:] = (1/N) * sum_{tgt : colMax==0} Wh[b,tgt,:]
__global__ void gat_uniform_cols(const float* __restrict__ colMax,
                                 const float* __restrict__ Wh,
                                 float* __restrict__ u) {
  __shared__ float acc[FOUT_];
  int b = blockIdx.x;
  for (int f = threadIdx.x; f < FOUT_; f += blockDim.x) acc[f] = 0.f;
  __syncthreads();
  float local[FOUT_];
  #pragma unroll
  for (int f = 0; f < FOUT_; ++f) local[f] = 0.f;
  for (int tgt = threadIdx.x; tgt < N_; tgt += blockDim.x) {
    if (colMax[b * N_ + tgt] == 0.f) {
      const float* row = Wh + ((size_t)b * N_ + tgt) * FOUT_;
      #pragma unroll
      for (int f = 0; f < FOUT_; ++f) local[f] += row[f];
    }
  }
  #pragma unroll
  for (int f = 0; f < FOUT_; ++f) atomicAdd(&acc[f], local[f]);
  __syncthreads();
  for (int f = threadIdx.x; f < FOUT_; f += blockDim.x)
    u[b * FOUT_ + f] = acc[f] * (1.0f / N_);
}

// Pass E: h_prime = accum + u (broadcast over rows), then ELU (alpha=1).
__global__ void gat_finalize(const float* __restrict__ accum,
                             const float* __restrict__ u,
                             float* __restrict__ out) {
  size_t gid = (size_t)blockIdx.x * blockDim.x + threadIdx.x;
  if (gid >= (size_t)B_ * N_ * FOUT_) return;
  int f = (int)(gid % FOUT_);
  int b = (int)(gid / ((size_t)N_ * FOUT_));
  float hp = accum[gid] + u[b * FOUT_ + f];
  out[gid] = hp > 0.f ? hp : expm1f(hp);
}

// ---------------------------------------------------------------------------
extern "C" void kernel_launch(void* const* d_in, const int* in_sizes, int n_in,
                              void* d_out, int out_size, void* d_ws, size_t ws_size,
                              hipStream_t stream) {
  const float* h    = (const float*)d_in[0];   // [B, N, FIN]
  const int*   edge = (const int*)d_in[1];     // [B, 2, E]
  const float* W    = (const float*)d_in[2];   // [FIN, FOUT]
  const float* a    = (const float*)d_in[3];   // [2*FOUT, 1]
  float* out        = (float*)d_out;           // [B, N, FOUT]

  // Workspace layout (256B-aligned carve-out)
  char* ws = (char*)d_ws;
  size_t off = 0;
  auto carve = [&](size_t bytes) {
    size_t o = off;
    off = (off + bytes + 255) & ~(size_t)255;
    return o;
  };
  const size_t whB     = (size_t)B_ * N_ * FOUT_ * sizeof(float);      // 4 MB
  const size_t accB    = whB;                                          // 4 MB
  const size_t sB      = (size_t)N_ * sizeof(float);                   // 16 KB
  const size_t colB    = (size_t)B_ * N_ * sizeof(float);              // 64 KB
  const size_t uB      = (size_t)B_ * FOUT_ * sizeof(float);           // 1 KB
  const size_t listB   = (size_t)B_ * E_ * sizeof(unsigned long long); // 2 MB
  const size_t bitmapB = ((size_t)B_ * N_ * N_ / 32) * sizeof(unsigned); // 8 MB

  float* Wh                = (float*)(ws + carve(whB));
  float* accum             = (float*)(ws + carve(accB));
  float* s1                = (float*)(ws + carve(sB));
  float* s2                = (float*)(ws + carve(sB));
  float* colMax            = (float*)(ws + carve(colB));
  float* colSum            = (float*)(ws + carve(colB));
  float* u                 = (float*)(ws + carve(uB));
  unsigned* cnt            = (unsigned*)(ws + carve(sizeof(unsigned)));
  unsigned long long* list = (unsigned long long*)(ws + carve(listB));
  unsigned* bitmap         = (unsigned*)(ws + carve(bitmapB));

  // Deterministic per-call init (graph-capture-safe memsets)
  hipMemsetAsync(accum,  0, accB,    stream);
  hipMemsetAsync(colMax, 0, colB,    stream);
  hipMemsetAsync(colSum, 0, colB,    stream);
  hipMemsetAsync(u,      0, uB,      stream);
  hipMemsetAsync(cnt,    0, sizeof(unsigned), stream);
  hipMemsetAsync(bitmap, 0, bitmapB, stream);

  // K1: Wh = h @ W (WMMA f32). 1024 M-tiles, 4 waves/block cover the 4 N-tiles.
  gat_wh_gemm<<<(B_ * N_) / 16, 128, 0, stream>>>(h, W, Wh);

  // K2: per-node score scalars from batch-0 rows.
  gat_node_scores<<<N_ / 256, 256, 0, stream>>>(Wh, a, s1, s2);

  // Sparse softmax + SpMM passes.
  gat_edge_pass_a<<<(B_ * E_) / 256, 256, 0, stream>>>(edge, s1, s2, bitmap,
                                                       colMax, cnt, list);
  gat_edge_pass_b<<<(B_ * E_) / 256, 256, 0, stream>>>(list, cnt, colMax, colSum);
  gat_edge_pass_c<<<((size_t)B_ * E_ * 64) / 256, 256, 0, stream>>>(
      list, cnt, colMax, colSum, Wh, accum);

  // Uniform-column rank-1 term, then finalize with ELU.
  gat_uniform_cols<<<B_, 256, 0, stream>>>(colMax, Wh, u);
  gat_finalize<<<((size_t)B_ * N_ * FOUT_) / 256, 256, 0, stream>>>(accum, u, out);
}